// FMClassifier_84146999263537
// MI455X (gfx1250) — compile-verified
//
#include <hip/hip_runtime.h>
#include <math.h>

// ---------------------------------------------------------------------------
// FM classifier: out = sigmoid( x @ [W_w ; triu(Vn Vn^T)] + W_b )
//   N_LIN=512, N_PAIRS=130816, D=131328, B=2048, K=16
// Stage 1: w[0:512]       = W_w                       (trivial copy)
// Stage 2: w[512:131328]  = triu(G), G = Vn Vn^T via V_WMMA_F32_16X16X4_F32
// Stage 3: out[b] = sigmoid( dot(x[b], w) + W_b )     (HBM-bound GEMV)
// ---------------------------------------------------------------------------

typedef float v2f __attribute__((ext_vector_type(2)));
typedef float v4f __attribute__((ext_vector_type(4)));
typedef float v8f __attribute__((ext_vector_type(8)));

#define FM_NLIN   512
#define FM_NPAIRS 130816
#define FM_D      (FM_NLIN + FM_NPAIRS)   // 131328 (multiple of 4)
#define FM_B      2048

// ---- Stage 1: linear weights into fused vector ----------------------------
__global__ void fm_copy_linear(const float* __restrict__ Ww,
                               float* __restrict__ w) {
    w[threadIdx.x] = Ww[threadIdx.x];     // 512 threads
}

// ---- Stage 2: pairwise weights via fp32 WMMA ------------------------------
// One wave (32 lanes) computes one 16x16 tile of G = Vn Vn^T by chaining
// four V_WMMA_F32_16X16X4_F32 ops over K=16. Only upper-triangular tiles run.
__global__ __launch_bounds__(32)
void fm_pair_weights(const float* __restrict__ V,   // row-major 512x16 used
                     float* __restrict__ w) {
    const int tj = blockIdx.x;            // column tile (N dim of D)
    const int ti = blockIdx.y;            // row tile    (M dim of D)
    if (tj < ti) return;                  // uniform: whole wave exits pre-WMMA

    const int lane = threadIdx.x;         // wave32
    const int half = lane >> 4;           // 0: lanes 0-15, 1: lanes 16-31
    const int l15  = lane & 15;

    // A fragment: A[m][k] = Vn[ti*16+m][k]; lane gives m, half selects K pair.
    // B fragment: B[k][n] = Vn[tj*16+n][k]; same load pattern (G symmetric).
    const float* Arow = V + (ti * 16 + l15) * 16;
    const float* Brow = V + (tj * 16 + l15) * 16;

    v8f c = {};
#pragma unroll
    for (int s = 0; s < 4; ++s) {
        const int kb = s * 4 + half * 2;  // even -> 8B aligned
        v2f a = *(const v2f*)(Arow + kb);
        v2f b = *(const v2f*)(Brow + kb);
        // 8 args: (neg_a, A, neg_b, B, c_mod, C, reuse_a, reuse_b)
        c = __builtin_amdgcn_wmma_f32_16x16x4_f32(
                false, a, false, b, (short)0, c, false, false);
    }

    // D layout (32-bit 16x16): VGPR r -> M = r + 8*half, N = l15.
    const int j = tj * 16 + l15;
#pragma unroll
    for (int r = 0; r < 8; ++r) {
        const int i = ti * 16 + r + half * 8;
        if (j > i) {
            // triu_indices(512, k=1) flat offset
            const int p = i * (FM_NLIN - 1) - (i * (i - 1)) / 2 + (j - i - 1);
            w[FM_NLIN + p] = c[r];
        }
    }
}

// ---- Stage 3: HBM-bound GEMV + sigmoid ------------------------------------
// One 256-thread block (8 waves) per row. x is streamed once -> non-temporal
// loads so the 513 KB shared w vector stays resident in L2/WGP$.
__global__ __launch_bounds__(256)
void fm_gemv(const float* __restrict__ x,
             const float* __restrict__ w,
             const float* __restrict__ Wb,
             float* __restrict__ out) {
    __shared__ float red[256];
    const int b   = blockIdx.x;
    const int tid = threadIdx.x;

    const v4f* xr = (const v4f*)(x + (size_t)b * FM_D);
    const v4f* wr = (const v4f*)w;

    v4f acc = {};
    const int nvec = FM_D / 4;            // 32832
    for (int idx = tid; idx < nvec; idx += 256) {
        v4f xv = __builtin_nontemporal_load(xr + idx);  // global_load_b128 NT
        v4f wv = wr[idx];                               // cached (RT)
        acc += xv * wv;
    }
    red[tid] = acc.x + acc.y + acc.z + acc.w;
    __syncthreads();
#pragma unroll
    for (int off = 128; off > 0; off >>= 1) {
        if (tid < off) red[tid] += red[tid + off];
        __syncthreads();
    }
    if (tid == 0) {
        const float z = red[0] + Wb[0];
        out[b] = 1.0f / (1.0f + __expf(-z));
    }
}

// ---------------------------------------------------------------------------
extern "C" void kernel_launch(void* const* d_in, const int* in_sizes, int n_in,
                              void* d_out, int out_size, void* d_ws, size_t ws_size,
                              hipStream_t stream) {
    const float* x  = (const float*)d_in[0];   // (2048, 131328)
    const float* Ww = (const float*)d_in[1];   // (1, 512)
    const float* Wb = (const float*)d_in[2];   // (1,)
    const float* V  = (const float*)d_in[3];   // (130816, 16), only [:512] used
    float* w   = (float*)d_ws;                 // fused weights: 131328 floats
    float* out = (float*)d_out;                // 2048 floats

    fm_copy_linear<<<1, FM_NLIN, 0, stream>>>(Ww, w);
    fm_pair_weights<<<dim3(32, 32), 32, 0, stream>>>(V, w);
    fm_gemv<<<FM_B, 256, 0, stream>>>(x, w, Wb, out);
}